// SegmentScore_72146860638313
// MI455X (gfx1250) — compile-verified
//
#include <hip/hip_runtime.h>
#include <hip/hip_bf16.h>
#include <math.h>

// Problem constants (from reference setup_inputs)
#define B_ 8
#define S_ 256
#define L_ 64
#define E_ 128
#define SCALE_ 0.08838834764831845f   // 1/sqrt(128)

typedef float v2f __attribute__((ext_vector_type(2)));
typedef float v8f __attribute__((ext_vector_type(8)));

// ---------------------------------------------------------------------------
// Stage 1: rawT[b, s, l] = sum_k harmony[b,l,k] * note[b,s,k]
// One wave (32 lanes) computes one 16(l) x 16(s) tile via 32 chained
// V_WMMA_F32_16X16X4_F32 ops (K=128 in steps of 4). f32 WMMA keeps full
// precision for f32 inputs. EXEC is all-ones (blockDim == 32).
//
// Fragment layouts (ISA 7.12.2, 32-bit 16x4 A):
//   A: lane -> M = lane&15 ; VGPR j holds K = (lane>>4)*2 + j
//   B: mirrored, lane -> N = lane&15 ; VGPR j holds K = (lane>>4)*2 + j
//   C/D: VGPR r -> M = r + 8*(lane>>4), N = lane&15
// ---------------------------------------------------------------------------
__global__ __launch_bounds__(32) void seg_wmma_gemm(const float* __restrict__ note,
                                                    const float* __restrict__ harmony,
                                                    float* __restrict__ rawT) {
  const int lane = threadIdx.x;          // 0..31
  const int s0 = blockIdx.x * 16;        // S tile
  const int l0 = blockIdx.y * 16;        // L tile
  const int b  = blockIdx.z;
  const int half = lane >> 4;            // 0 or 1
  const int lidx = lane & 15;            // 0..15

  const float* Arow = harmony + ((size_t)b * L_ + (l0 + lidx)) * E_;  // A row (M dim)
  const float* Brow = note    + ((size_t)b * S_ + (s0 + lidx)) * E_;  // B col (N dim), B[k][n]=note[n][k]

  v8f c = {};
#pragma unroll 4
  for (int kk = 0; kk < E_; kk += 4) {
    const int kb = kk + (half << 1);
    v2f a, bf;
    a[0]  = Arow[kb + 0];
    a[1]  = Arow[kb + 1];
    bf[0] = Brow[kb + 0];
    bf[1] = Brow[kb + 1];
    // D = A(16x4) * B(4x16) + C ; 8 args: neg_a, A, neg_b, B, c_mod, C, reuse_a, reuse_b
    c = __builtin_amdgcn_wmma_f32_16x16x4_f32(false, a, false, bf, (short)0, c, false, false);
  }

#pragma unroll
  for (int r = 0; r < 8; ++r) {
    const int l = l0 + r + (half << 3);
    const int s = s0 + lidx;
    rawT[((size_t)b * S_ + s) * L_ + l] = c[r];
  }
}

// ---------------------------------------------------------------------------
// Stage 2: suffT[b, s, l] = max_{t >= s} rawT[b,t,l] * scale  (suffix max of
// scaled logits, matching the reference's finite per-start max).
// 64 threads = l, coalesced [*, l] rows; data is L2-resident (512 KB).
// ---------------------------------------------------------------------------
__global__ __launch_bounds__(64) void seg_suffix_max(const float* __restrict__ rawT,
                                                     float* __restrict__ suffT) {
  const int l = threadIdx.x;
  const int b = blockIdx.x;
  float m = -INFINITY;
  for (int s = S_ - 1; s >= 0; --s) {
    const size_t idx = ((size_t)b * S_ + s) * L_ + l;
    const float v = rawT[idx] * SCALE_;
    m = fmaxf(m, v);
    suffT[idx] = m;
  }
}

// ---------------------------------------------------------------------------
// Stage 3: for each (b, start): running softmax-weighted average over end.
//   t   = exp(raw*scale - m_start)
//   c1 += t ; c2 += t*raw ; out[b,start,end,l] = c1>0 ? c2*rcp(c1) : 0
//
// The end < start region of a (b,start) slab is a single contiguous span of
// start*64 floats -> filled with float4 (global_store_b128) stores.
// Live region: block = (b,start), 64 threads = l -> 256 B contiguous b32
// stores per end. rcp via v_rcp_f32 (1 trans op) instead of the full-rate
// divide expansion: per-element chain is exp + rcp + 2 FMA, below the
// 134 MB / 23.3 TB/s ~ 6 us store floor. All rawT reads hit L2 (512 KB).
// ---------------------------------------------------------------------------
__global__ __launch_bounds__(64) void seg_scan(const float* __restrict__ rawT,
                                               const float* __restrict__ suffT,
                                               float* __restrict__ out) {
  const int tid   = threadIdx.x;
  const int start = blockIdx.x;
  const int b     = blockIdx.y;

  float* slab = out + ((size_t)b * S_ + start) * S_ * L_;  // [S_end, L] slab

  // Zero-fill rows end = 0..start-1 : contiguous start*64 floats, b128 stores.
  {
    float4* z = (float4*)slab;
    const int nvec = start * (L_ / 4);   // start * 16 float4s
    const float4 zero4 = make_float4(0.f, 0.f, 0.f, 0.f);
    for (int i = tid; i < nvec; i += 64)
      z[i] = zero4;
  }

  // Running softmax-weighted scan for end = start..S-1.
  const int l = tid;
  const float m = suffT[((size_t)b * S_ + start) * L_ + l];
  float* orow = slab + l;

  float c1 = 0.0f, c2 = 0.0f;
  for (int end = start; end < S_; ++end) {
    const float v = rawT[((size_t)b * S_ + end) * L_ + l];
    const float t = __expf(__fmaf_rn(v, SCALE_, -m));
    c1 += t;
    c2 = __fmaf_rn(t, v, c2);
    const float r = __builtin_amdgcn_rcpf(c1);          // v_rcp_f32
    orow[(size_t)end * L_] = (c1 > 0.0f) ? (c2 * r) : 0.0f;
  }
}

extern "C" void kernel_launch(void* const* d_in, const int* in_sizes, int n_in,
                              void* d_out, int out_size, void* d_ws, size_t ws_size,
                              hipStream_t stream) {
  const float* note    = (const float*)d_in[0];  // [B, S, E]
  const float* harmony = (const float*)d_in[1];  // [B, L, E]
  float* out = (float*)d_out;                    // [B, S, S, L]

  float* rawT  = (float*)d_ws;                   // [B, S, L]  512 KB
  float* suffT = rawT + (size_t)B_ * S_ * L_;    // [B, S, L]  512 KB

  (void)in_sizes; (void)n_in; (void)out_size; (void)ws_size;

  seg_wmma_gemm<<<dim3(S_ / 16, L_ / 16, B_), 32, 0, stream>>>(note, harmony, rawT);
  seg_suffix_max<<<dim3(B_), 64, 0, stream>>>(rawT, suffT);
  seg_scan<<<dim3(S_, B_), 64, 0, stream>>>(rawT, suffT, out);
}